// Label_Decoder_Global_Infor_24232205484690
// MI455X (gfx1250) — compile-verified
//
#include <hip/hip_runtime.h>
#include <math.h>

// ---------------- problem constants ----------------
#define FEA      512
#define LAB_BOS  402
#define HALF_LAB 200
#define KP1      21
#define L_OBS    100
#define K_LIVE   (L_OBS * FEA)     // 51200 live rows of W_all (rest multiply zeros)
#define BIGCH    256               // k-rows per block in the big GEMV
#define NCHUNK   (K_LIVE / BIGCH)  // 200 partial chunks
#define NGATE    (4 * FEA)         // 2048

typedef __attribute__((ext_vector_type(2))) float v2f;
typedef __attribute__((ext_vector_type(8))) float v8f;

// ---------------------------------------------------------------------------
// Per-wave fp32 WMMA GEMV tile:  out[nbase..nbase+15] (+)= x[0..K) . W[0..K, n]
// W is [K x ldN] row-major.  Mapping (ISA 7.12.2, 32-bit matrices, wave32):
//   A (16x4):  lane m(0..15)=M, VGPR0 holds K=0|2 (lane halves), VGPR1 K=1|3
//              -> A[m][k'] = W[(kbase+k')*ldN + nbase + m]   (weight tile)
//   B (4x16):  row k' constant = x[kbase+k']  (broadcast along N, so the
//              result is replicated across all 16 columns of D)
//   C/D (16x16): VGPR v, lanes 0-15 -> M=v, lanes 16-31 -> M=v+8
// ---------------------------------------------------------------------------
__device__ __forceinline__ v8f wmma_gemv_acc(v8f acc,
                                             const float* __restrict__ W, int ldN,
                                             const float* __restrict__ x, int K,
                                             int nbase, int Nvalid) {
  const int lane = threadIdx.x & 31;
  const int m    = lane & 15;
  const int hf   = lane >> 4;           // lane half selects K offset 0 / 2
  const int n    = nbase + m;
  const int nc   = (n < Nvalid) ? n : (Nvalid - 1);
  const float mk = (n < Nvalid) ? 1.0f : 0.0f;   // tail mask without divergence
  for (int k = 0; k < K; k += 4) {
    const int k0 = k + 2 * hf;
    v2f a, b;
    a.x = W[(size_t)k0 * ldN + nc] * mk;
    a.y = W[(size_t)(k0 + 1) * ldN + nc] * mk;
    b.x = x[k0];
    b.y = x[k0 + 1];
    acc = __builtin_amdgcn_wmma_f32_16x16x4_f32(false, a, false, b,
                                                (short)0, acc, false, false);
  }
  return acc;
}

__device__ __forceinline__ void wmma_gemv_store(v8f acc, const float* __restrict__ bias,
                                                float* __restrict__ out,
                                                float* __restrict__ out2,
                                                int nbase, int Nvalid, int relu) {
  const int lane = threadIdx.x & 31;
  const int hf   = lane >> 4;
  if ((lane & 15) == 0) {               // lanes 0 and 16 hold M = v and v+8
#pragma unroll
    for (int v = 0; v < 8; ++v) {
      const int n = nbase + v + 8 * hf;
      if (n < Nvalid) {
        float r = acc[v] + bias[n];
        if (relu) r = fmaxf(r, 0.0f);
        out[n] = r;
        if (out2) out2[n] = r;
      }
    }
  }
}

// Generic WMMA GEMV: out[n] = (relu?)(x @ W + bias), one 16-wide tile per wave.
__global__ void k_wmma_gemv(const float* __restrict__ W, const float* __restrict__ x,
                            const float* __restrict__ bias, float* __restrict__ out,
                            float* __restrict__ out2, int K, int ldN, int Nvalid,
                            int relu) {
  const int wave  = (int)((blockIdx.x * blockDim.x + threadIdx.x) >> 5);
  const int nbase = wave * 16;
  if (nbase >= Nvalid) return;          // wave-uniform exit (EXEC stays all-1s)
  v8f acc = {};
  acc = wmma_gemv_acc(acc, W, ldN, x, K, nbase, Nvalid);
  wmma_gemv_store(acc, bias, out, out2, nbase, Nvalid, relu);
}

// LSTM gates: gates = x@W_ih + h@W_hh + b_ih + b_hh   (N = 2048, two K=512 passes)
__global__ void k_wmma_gates(const float* __restrict__ W_ih, const float* __restrict__ W_hh,
                             const float* __restrict__ x, const float* __restrict__ h,
                             const float* __restrict__ b_ih, const float* __restrict__ b_hh,
                             float* __restrict__ gates) {
  const int wave  = (int)((blockIdx.x * blockDim.x + threadIdx.x) >> 5);
  const int nbase = wave * 16;
  if (nbase >= NGATE) return;
  v8f acc = {};
  acc = wmma_gemv_acc(acc, W_ih, NGATE, x, FEA, nbase, NGATE);
  acc = wmma_gemv_acc(acc, W_hh, NGATE, h, FEA, nbase, NGATE);
  const int lane = threadIdx.x & 31;
  const int hf   = lane >> 4;
  if ((lane & 15) == 0) {
#pragma unroll
    for (int v = 0; v < 8; ++v) {
      const int n = nbase + v + 8 * hf;
      gates[n] = acc[v] + b_ih[n] + b_hh[n];
    }
  }
}

// ---------------------------------------------------------------------------
// Stage 1: gather/scale preparation
//  blocks 0..199 : flat[e] = lab_table[(2*i + 1 - lab_value[i])*FEA + j]
//  block 200     : vec1536[512+n]  = lab_table[(index+2)*FEA+n] * alpha
//  block 201     : vec1536[1024+n] = lab_table[(index+3)*FEA+n] * (1-alpha)
//  block 202     : cat_eh[512+n]   = h0[n]
// ---------------------------------------------------------------------------
__global__ void k_prepare(const int* __restrict__ index_p, const int* __restrict__ lab_value,
                          const float* __restrict__ alpha_p, const float* __restrict__ h0,
                          const float* __restrict__ lab_table,
                          float* __restrict__ flat, float* __restrict__ vec1536,
                          float* __restrict__ cat_eh) {
  const int bid = blockIdx.x, tid = threadIdx.x;
  if (bid < NCHUNK) {                             // 200 * 256 == 51200
    const int e = bid * 256 + tid;
    const int i = e >> 9, j = e & (FEA - 1);
    const int row = 2 * i + 1 - lab_value[i];
    flat[e] = lab_table[(size_t)row * FEA + j];
  } else if (bid == NCHUNK) {
    const float a = alpha_p[0];
    const int idx = index_p[0];
    for (int n = tid; n < FEA; n += 256)
      vec1536[FEA + n] = lab_table[(size_t)(idx + 2) * FEA + n] * a;
  } else if (bid == NCHUNK + 1) {
    const float a = 1.0f - alpha_p[0];
    const int idx = index_p[0];
    for (int n = tid; n < FEA; n += 256)
      vec1536[2 * FEA + n] = lab_table[(size_t)(idx + 3) * FEA + n] * a;
  } else {
    for (int n = tid; n < FEA; n += 256) cat_eh[FEA + n] = h0[n];
  }
}

// ---------------------------------------------------------------------------
// Stage 2: dominant GEMV over the live half of W_all (105 MB streamed once).
// One block = 256 k-rows; thread n (0..511) accumulates its output column over
// the chunk with perfectly coalesced 2KB row reads; deterministic partials.
// ---------------------------------------------------------------------------
__global__ void __launch_bounds__(FEA) k_gemv_big(const float* __restrict__ W_all,
                                                  const float* __restrict__ flat,
                                                  float* __restrict__ partials) {
  __shared__ float sf[BIGCH];
  const int kbase = blockIdx.x * BIGCH;
  for (int i = threadIdx.x; i < BIGCH; i += FEA) sf[i] = flat[kbase + i];
  __syncthreads();
  const int n = threadIdx.x;
  const float* Wp = W_all + (size_t)kbase * FEA + n;
  float acc = 0.0f;
#pragma unroll 8
  for (int kk = 0; kk < BIGCH; ++kk)
    acc = fmaf(sf[kk], __builtin_nontemporal_load(Wp + (size_t)kk * FEA), acc);
  partials[(size_t)blockIdx.x * FEA + n] = acc;
}

// Stage 3: fixed-order chunk reduction -> emb1 (into vec1536[0:512])
__global__ void k_reduce(const float* __restrict__ partials, const float* __restrict__ b_all,
                         float* __restrict__ vec1536) {
  const int n = threadIdx.x;
  float acc = b_all[n];
  for (int c = 0; c < NCHUNK; ++c) acc += partials[(size_t)c * FEA + n];
  vec1536[n] = acc;
}

// Attention: aw = inp @ W_aw + b_aw (21 outs, warp per output), then
// ctx[j] = sum_p aw[p] * enc[p*1024 + j]   (single block, 1024 threads)
__global__ void k_attn(const float* __restrict__ inp, const float* __restrict__ W_aw,
                       const float* __restrict__ b_aw, const float* __restrict__ enc,
                       float* __restrict__ ctx) {
  __shared__ float saw[32];
  const int warp = threadIdx.x >> 5, lane = threadIdx.x & 31;
  if (warp < KP1) {
    float s = 0.0f;
    for (int k = lane; k < FEA; k += 32)
      s = fmaf(inp[k], W_aw[(size_t)k * KP1 + warp], s);
#pragma unroll
    for (int o = 16; o > 0; o >>= 1) s += __shfl_down(s, o, 32);
    if (lane == 0) saw[warp] = s + b_aw[warp];
  }
  __syncthreads();
  const int j = threadIdx.x;                       // 1024 = 2*FEA
  float c = 0.0f;
#pragma unroll
  for (int p = 0; p < KP1; ++p) c = fmaf(saw[p], enc[(size_t)p * 2 * FEA + j], c);
  ctx[j] = c;
}

// LSTM pointwise (torch gate order i,f,g,o); writes h_new/c_new to d_out too.
__global__ void k_lstm(const float* __restrict__ gates, const float* __restrict__ c0,
                       float* __restrict__ out, float* __restrict__ hbuf) {
  const int n = threadIdx.x;
  const float ig = gates[n], fg = gates[FEA + n];
  const float gg = gates[2 * FEA + n], og = gates[3 * FEA + n];
  const float si = 1.0f / (1.0f + expf(-ig));
  const float sf = 1.0f / (1.0f + expf(-fg));
  const float so = 1.0f / (1.0f + expf(-og));
  const float cn = sf * c0[n] + si * tanhf(gg);
  const float hn = so * tanhf(cn);
  out[LAB_BOS + n]       = hn;   // h_new at offset 402
  out[LAB_BOS + FEA + n] = cn;   // c_new at offset 914
  hbuf[n] = hn;
}

// log-softmax over 402 logits, deterministic tree reductions.
__global__ void k_logsoftmax(const float* __restrict__ logits, float* __restrict__ out) {
  __shared__ float red[FEA];
  const int t = threadIdx.x;
  const float v = (t < LAB_BOS) ? logits[t] : -INFINITY;
  red[t] = v;
  __syncthreads();
  for (int s = FEA / 2; s > 0; s >>= 1) {
    if (t < s) red[t] = fmaxf(red[t], red[t + s]);
    __syncthreads();
  }
  const float mx = red[0];
  __syncthreads();
  red[t] = (t < LAB_BOS) ? expf(v - mx) : 0.0f;
  __syncthreads();
  for (int s = FEA / 2; s > 0; s >>= 1) {
    if (t < s) red[t] += red[t + s];
    __syncthreads();
  }
  const float lse = logf(red[0]);
  if (t < LAB_BOS) out[t] = v - mx - lse;
}

// ---------------------------------------------------------------------------
extern "C" void kernel_launch(void* const* d_in, const int* in_sizes, int n_in,
                              void* d_out, int out_size, void* d_ws, size_t ws_size,
                              hipStream_t stream) {
  (void)in_sizes; (void)n_in; (void)out_size; (void)ws_size;
  const int*   index_p   = (const int*)  d_in[0];
  const int*   lab_value = (const int*)  d_in[1];
  const float* alpha_p   = (const float*)d_in[2];
  const float* h0        = (const float*)d_in[3];
  const float* c0        = (const float*)d_in[4];
  const float* enc       = (const float*)d_in[5];
  const float* lab_table = (const float*)d_in[6];
  const float* W_all     = (const float*)d_in[7];
  const float* b_all     = (const float*)d_in[8];
  const float* W3        = (const float*)d_in[9];
  const float* b3        = (const float*)d_in[10];
  const float* W_eh      = (const float*)d_in[11];
  const float* b_eh      = (const float*)d_in[12];
  const float* W_aw      = (const float*)d_in[13];
  const float* b_aw      = (const float*)d_in[14];
  const float* W_lin     = (const float*)d_in[15];
  const float* b_lin     = (const float*)d_in[16];
  const float* W_ih      = (const float*)d_in[17];
  const float* W_hh      = (const float*)d_in[18];
  const float* b_ih      = (const float*)d_in[19];
  const float* b_hh      = (const float*)d_in[20];
  const float* W_out     = (const float*)d_in[21];
  const float* b_out     = (const float*)d_in[22];
  float* out = (float*)d_out;

  // workspace layout (floats); total ~650 KB
  float* ws       = (float*)d_ws;
  float* flat     = ws;                       // 51200
  float* partials = flat + K_LIVE;            // 200*512 = 102400
  float* vec1536  = partials + NCHUNK * FEA;  // 1536  [emb1|emb2|emb3]
  float* cat_eh   = vec1536 + 3 * FEA;        // 1024  [emb|h]
  float* inp      = cat_eh + 2 * FEA;         // 512
  float* ctx      = inp + FEA;                // 1024
  float* cat2     = ctx + 2 * FEA;            // 1024  [emb|app]
  float* xbuf     = cat2 + 2 * FEA;           // 512
  float* gates    = xbuf + FEA;               // 2048
  float* hbuf     = gates + NGATE;            // 512
  float* logits   = hbuf + FEA;               // 512 (402 used)

  // 1) gathers / scalar-scaled embeddings / h copy
  k_prepare<<<NCHUNK + 3, 256, 0, stream>>>(index_p, lab_value, alpha_p, h0,
                                            lab_table, flat, vec1536, cat_eh);
  // 2) dominant 51200x512 GEMV (HBM-bound, ~105 MB streamed non-temporally)
  k_gemv_big<<<NCHUNK, FEA, 0, stream>>>(W_all, flat, partials);
  // 3) deterministic reduction -> emb1
  k_reduce<<<1, FEA, 0, stream>>>(partials, b_all, vec1536);
  // 4) emb = vec1536 @ W3 + b3   (also feeds cat2[0:512])
  k_wmma_gemv<<<4, 256, 0, stream>>>(W3, vec1536, b3, cat_eh, cat2, 3 * FEA, FEA, FEA, 0);
  // 5) inp = [emb|h] @ W_eh + b_eh
  k_wmma_gemv<<<4, 256, 0, stream>>>(W_eh, cat_eh, b_eh, inp, nullptr, 2 * FEA, FEA, FEA, 0);
  // 6) aw = inp @ W_aw + b_aw ; ctx = aw @ enc
  k_attn<<<1, 1024, 0, stream>>>(inp, W_aw, b_aw, enc, ctx);
  // 7) app = ctx @ W_lin + b_lin  -> cat2[512:1024]
  k_wmma_gemv<<<4, 256, 0, stream>>>(W_lin, ctx, b_lin, cat2 + FEA, nullptr, 2 * FEA, FEA, FEA, 0);
  // 8) x = relu([emb|app] @ W_lin + b_lin)
  k_wmma_gemv<<<4, 256, 0, stream>>>(W_lin, cat2, b_lin, xbuf, nullptr, 2 * FEA, FEA, FEA, 1);
  // 9) gates = x@W_ih + h@W_hh + biases   (128 tiles -> 16 blocks x 8 waves)
  k_wmma_gates<<<16, 256, 0, stream>>>(W_ih, W_hh, xbuf, cat_eh + FEA, b_ih, b_hh, gates);
  // 10) LSTM cell pointwise -> h_new/c_new (written straight into d_out)
  k_lstm<<<1, FEA, 0, stream>>>(gates, c0, out, hbuf);
  // 11) logits = h_new @ W_out + b_out  (26 tiles, tail-masked)
  k_wmma_gemv<<<4, 256, 0, stream>>>(W_out, hbuf, b_out, logits, nullptr, FEA, LAB_BOS, LAB_BOS, 0);
  // 12) log-softmax -> d_out[0:402]
  k_logsoftmax<<<1, FEA, 0, stream>>>(logits, out);
}